// MyModel_40621800686195
// MI455X (gfx1250) — compile-verified
//
#include <hip/hip_runtime.h>
#include <math.h>

// ---- problem constants (from reference) ----
#define BN    512
#define INF_  127            // IN
#define L1_   32
#define L2_   31
#define DIMV  2
#define TPTS  63             // T
#define KPATH 64             // K
#define NCLS_ 10
#define NMEAN 62             // L2*DIM
#define MATEL 795
#define NTOT  857            // NMEAN + MATEL
#define SIGD  120            // 3+9+27+81
#define MSTRD 66             // LDS row stride for M tile (bank padding)
#define ESTRD 65             // LDS row stride for eps tile (bank padding)

typedef float v2f __attribute__((ext_vector_type(2)));
typedef float v8f __attribute__((ext_vector_type(8)));

// ============================================================
// Kernel 1: Y[512,857] = x[512,127] @ [W_mean;W_cov]^T + bias
// WMMA f32 16x16x4. 32 M-tiles x 54 N-tiles = 1728 tiles,
// 8 waves/block -> 216 blocks. All selects hoisted; K-tail peeled.
// ============================================================
__global__ __launch_bounds__(256)
void k_gemm_xw(const float* __restrict__ x,
               const float* __restrict__ Wm, const float* __restrict__ bm,
               const float* __restrict__ Wc, const float* __restrict__ bc,
               float* __restrict__ Y) {
    const int lane = threadIdx.x & 31;
    const int wave = threadIdx.x >> 5;
    const int tile = blockIdx.x * 8 + wave;          // 0..1727
    const int nt = tile % 54, mt = tile / 54;
    const int m0 = mt * 16, n0 = nt * 16;
    const int row = lane & 15;
    const int grp = lane >> 4;                       // 0 or 1
    const int n   = n0 + row;                        // output column
    const bool nok = (n < NTOT);
    const int ncl = nok ? n : (NTOT - 1);            // clamp: OOR lanes read valid mem

    const float* __restrict__ wrow =
        (ncl < NMEAN) ? (Wm + ncl * INF_) : (Wc + (ncl - NMEAN) * INF_);
    const float* __restrict__ arow = x + (m0 + row) * INF_;

    v8f c = {};
    #pragma unroll 4
    for (int k0 = 0; k0 < 124; k0 += 4) {            // 31 unguarded steps
        const int kk = k0 + grp * 2;
        __builtin_prefetch(wrow + kk + 16, 0, 3);    // global_prefetch_b8
        v2f a, b;
        a[0] = arow[kk];     a[1] = arow[kk + 1];
        b[0] = wrow[kk];     b[1] = wrow[kk + 1];
        c = __builtin_amdgcn_wmma_f32_16x16x4_f32(false, a, false, b,
                                                  (short)0, c, false, false);
    }
    {   // tail: k0 = 124, kk in {124,126}, kk+1 in {125,127}
        const int kk = 124 + grp * 2;
        v2f a, b;
        a[0] = arow[kk];
        a[1] = (kk + 1 < INF_) ? arow[kk + 1] : 0.f;
        b[0] = wrow[kk];
        b[1] = (kk + 1 < INF_) ? wrow[kk + 1] : 0.f;
        c = __builtin_amdgcn_wmma_f32_16x16x4_f32(false, a, false, b,
                                                  (short)0, c, false, false);
    }
    if (nok) {
        const float bias = (n < NMEAN) ? bm[n] : bc[n - NMEAN];
        #pragma unroll
        for (int r = 0; r < 8; ++r) {
            const int rowM = m0 + r + 8 * grp;
            Y[rowM * NTOT + n] = c[r] + bias;
        }
    }
}

// ============================================================
// Kernel 2 (fused scatter + batched GEMM):
//   NV[b] = scatter(sc[b]) (62x62, padded 64) @ eps[b] (62x64) + mean
// One block per batch, 16 waves, M & eps staged in bank-padded LDS.
// ============================================================
__global__ __launch_bounds__(512)
void k_bgemm(const float* __restrict__ Y,
             const int* __restrict__ xi, const int* __restrict__ yi,
             const float* __restrict__ eps, float* __restrict__ NV) {
    __shared__ float ldsM[64 * MSTRD];
    __shared__ float ldsE[64 * ESTRD];
    const int b = blockIdx.x;
    const int tid = threadIdx.x;

    // zero M tile (cols 0..63 only are ever read) + stage padded eps
    const float* __restrict__ eb = eps + (size_t)b * NMEAN * KPATH;
    for (int i = tid; i < 64 * 64; i += 512) {
        const int r = i >> 6, cc = i & 63;
        ldsM[r * MSTRD + cc] = 0.f;
        ldsE[r * ESTRD + cc] = (r < NMEAN) ? eb[r * KPATH + cc] : 0.f;
    }
    __syncthreads();
    // scatter banded sqrt-cov entries into LDS M tile
    const float* __restrict__ sc = Y + (size_t)b * NTOT + NMEAN;
    for (int e = tid; e < MATEL; e += 512)
        ldsM[xi[e] * MSTRD + yi[e]] = sc[e];
    __syncthreads();

    const int lane = tid & 31;
    const int wave = tid >> 5;                       // 0..15
    const int mt = wave >> 2, nt = wave & 3;
    const int m0 = mt * 16, n0 = nt * 16;
    const int row = lane & 15;
    const int grp = lane >> 4;
    const int col = n0 + row;                        // path index k (0..63)

    v8f c = {};
    #pragma unroll
    for (int k0 = 0; k0 < 64; k0 += 4) {             // zeros pad k=62..63
        const int kk = k0 + grp * 2;
        v2f a, bf;
        a[0]  = ldsM[(m0 + row) * MSTRD + kk];
        a[1]  = ldsM[(m0 + row) * MSTRD + kk + 1];
        bf[0] = ldsE[kk * ESTRD + col];
        bf[1] = ldsE[(kk + 1) * ESTRD + col];
        c = __builtin_amdgcn_wmma_f32_16x16x4_f32(false, a, false, bf,
                                                  (short)0, c, false, false);
    }
    float* __restrict__ nvb = NV + (size_t)b * 64 * KPATH;
    #pragma unroll
    for (int r = 0; r < 8; ++r) {
        const int rowM = m0 + r + 8 * grp;
        const float mval = (rowM < NMEAN) ? Y[(size_t)b * NTOT + rowM] : 0.f;
        nvb[rowM * KPATH + col] = c[r] + mval;
    }
}

// ============================================================
// Kernel 3: per-(b,k) truncated signature (level 4, D=3) via
// Chen's relation; mean over K=64 via LDS float atomics.
// ============================================================
__device__ __forceinline__ void load_point(const float* __restrict__ xrow,
                                           const float* __restrict__ nvb,
                                           int k, int t, float p[3]) {
    const int j = t >> 1;
    if ((t & 1) == 0) {
        p[0] = xrow[L1_ * DIMV + j];                 // known time
        p[1] = xrow[2 * j];
        p[2] = xrow[2 * j + 1];
    } else {
        p[0] = xrow[L1_ * DIMV + L1_ + j];           // sampled time
        p[1] = nvb[(2 * j) * KPATH + k];
        p[2] = nvb[(2 * j + 1) * KPATH + k];
    }
}

__global__ __launch_bounds__(64)
void k_signature(const float* __restrict__ x, const float* __restrict__ NV,
                 float* __restrict__ MS) {
    const int b = blockIdx.x;
    const int k = threadIdx.x;                       // 0..63
    const float* xrow = x + b * INF_;
    const float* nvb  = NV + (size_t)b * 64 * KPATH;

    float s1[3] = {}, s2[9] = {}, s3[27] = {}, s4[81] = {};
    float prev[3], cur[3];
    load_point(xrow, nvb, k, 0, prev);

    for (int t = 1; t < TPTS; ++t) {
        load_point(xrow, nvb, k, t, cur);
        float dx[3];
        #pragma unroll
        for (int d = 0; d < 3; ++d) { dx[d] = cur[d] - prev[d]; prev[d] = cur[d]; }

        float e2[9], e3[27];
        #pragma unroll
        for (int i = 0; i < 3; ++i)
            #pragma unroll
            for (int j = 0; j < 3; ++j) e2[i * 3 + j] = 0.5f * dx[i] * dx[j];
        #pragma unroll
        for (int i = 0; i < 9; ++i)
            #pragma unroll
            for (int j = 0; j < 3; ++j) e3[i * 3 + j] = (1.f / 3.f) * e2[i] * dx[j];

        // level 4 (uses OLD s1..s3)
        #pragma unroll
        for (int i = 0; i < 27; ++i)
            #pragma unroll
            for (int j = 0; j < 3; ++j) s4[i * 3 + j] += 0.25f * e3[i] * dx[j]; // e4
        #pragma unroll
        for (int i = 0; i < 3; ++i)
            #pragma unroll
            for (int j = 0; j < 27; ++j) s4[i * 27 + j] += s1[i] * e3[j];
        #pragma unroll
        for (int i = 0; i < 9; ++i)
            #pragma unroll
            for (int j = 0; j < 9; ++j) s4[i * 9 + j] += s2[i] * e2[j];
        #pragma unroll
        for (int i = 0; i < 27; ++i)
            #pragma unroll
            for (int j = 0; j < 3; ++j) s4[i * 3 + j] += s3[i] * dx[j];
        // level 3 (uses OLD s1, s2)
        #pragma unroll
        for (int n = 0; n < 27; ++n) s3[n] += e3[n];
        #pragma unroll
        for (int i = 0; i < 3; ++i)
            #pragma unroll
            for (int j = 0; j < 9; ++j) s3[i * 9 + j] += s1[i] * e2[j];
        #pragma unroll
        for (int i = 0; i < 9; ++i)
            #pragma unroll
            for (int j = 0; j < 3; ++j) s3[i * 3 + j] += s2[i] * dx[j];
        // level 2 (uses OLD s1)
        #pragma unroll
        for (int n = 0; n < 9; ++n) s2[n] += e2[n];
        #pragma unroll
        for (int i = 0; i < 3; ++i)
            #pragma unroll
            for (int j = 0; j < 3; ++j) s2[i * 3 + j] += s1[i] * dx[j];
        // level 1
        #pragma unroll
        for (int n = 0; n < 3; ++n) s1[n] += dx[n];
    }

    __shared__ float red[SIGD];
    for (int i = threadIdx.x; i < SIGD; i += blockDim.x) red[i] = 0.f;
    __syncthreads();
    #pragma unroll
    for (int n = 0; n < 3; ++n)  atomicAdd(&red[n], s1[n]);
    #pragma unroll
    for (int n = 0; n < 9; ++n)  atomicAdd(&red[3 + n], s2[n]);
    #pragma unroll
    for (int n = 0; n < 27; ++n) atomicAdd(&red[12 + n], s3[n]);
    #pragma unroll
    for (int n = 0; n < 81; ++n) atomicAdd(&red[39 + n], s4[n]);
    __syncthreads();
    for (int i = threadIdx.x; i < SIGD; i += blockDim.x)
        MS[b * SIGD + i] = red[i] * (1.0f / (float)KPATH);
}

// ============================================================
// Kernel 4: logits + log_softmax, one thread per batch row
// ============================================================
__global__ __launch_bounds__(256)
void k_head(const float* __restrict__ MS, const float* __restrict__ Wf,
            const float* __restrict__ bf, float* __restrict__ out) {
    const int b = blockIdx.x * blockDim.x + threadIdx.x;
    if (b >= BN) return;
    float l[NCLS_];
    #pragma unroll
    for (int c = 0; c < NCLS_; ++c) {
        float acc = bf[c];
        for (int i = 0; i < SIGD; ++i) acc += MS[b * SIGD + i] * Wf[c * SIGD + i];
        l[c] = acc;
    }
    float mx = l[0];
    #pragma unroll
    for (int c = 1; c < NCLS_; ++c) mx = fmaxf(mx, l[c]);
    float sum = 0.f;
    #pragma unroll
    for (int c = 0; c < NCLS_; ++c) sum += __expf(l[c] - mx);
    const float lse = mx + __logf(sum);
    #pragma unroll
    for (int c = 0; c < NCLS_; ++c) out[b * NCLS_ + c] = l[c] - lse;
}

// ============================================================
extern "C" void kernel_launch(void* const* d_in, const int* in_sizes, int n_in,
                              void* d_out, int out_size, void* d_ws, size_t ws_size,
                              hipStream_t stream) {
    const float* x   = (const float*)d_in[0];
    const float* Wm  = (const float*)d_in[1];
    const float* bm  = (const float*)d_in[2];
    const float* Wc  = (const float*)d_in[3];
    const float* bc  = (const float*)d_in[4];
    const float* Wf  = (const float*)d_in[5];
    const float* bfi = (const float*)d_in[6];
    const float* eps = (const float*)d_in[7];
    const int*   xi  = (const int*)d_in[8];
    const int*   yi  = (const int*)d_in[9];
    // d_in[10] (order) / d_in[11] (ext_order) are fixed patterns, hardcoded.

    float* ws = (float*)d_ws;
    float* Y  = ws;                                  // [512, 857]
    float* NV = Y  + (size_t)BN * NTOT;              // [512, 64, 64]
    float* MS = NV + (size_t)BN * 64 * KPATH;        // [512, 120]

    k_gemm_xw  <<<216, 256, 0, stream>>>(x, Wm, bm, Wc, bc, Y);
    k_bgemm    <<<BN, 512, 0, stream>>>(Y, xi, yi, eps, NV);
    k_signature<<<BN, 64, 0, stream>>>(x, NV, MS);
    k_head     <<<(BN + 255) / 256, 256, 0, stream>>>(MS, Wf, bfi, (float*)d_out);
}